// FraudGNN_83245056131912
// MI455X (gfx1250) — compile-verified
//
#include <hip/hip_runtime.h>
#include <hip/hip_bf16.h>

typedef __attribute__((ext_vector_type(2))) float v2f;
typedef __attribute__((ext_vector_type(8))) float v8f;

// ---------------------------------------------------------------------------
// Weight transpose: in [K, NOUT] row-major -> out [NOUT, K] row-major.
// Makes the WMMA B-fragment {B[k][col], B[k+1][col]} one contiguous float2.
// ---------------------------------------------------------------------------
__global__ void transpose_kernel(const float* __restrict__ in, float* __restrict__ out,
                                 int K, int NOUT) {
    const int i = blockIdx.x * blockDim.x + threadIdx.x;
    if (i < K * NOUT) {
        const int k = i / NOUT;
        const int n = i - k * NOUT;
        out[(long)n * K + k] = in[i];
    }
}

// ---------------------------------------------------------------------------
// Row-block fused GEMM on V_WMMA_F32_16X16X4_F32:
//   out = relu( [scale_rows(A1) @ B1] + [A2 @ B2 (opt)] + bias )
// One wave32 computes 16 rows x (TILES_N*16) cols. K is a compile-time
// constant so each column tile's displacement (tn*16*K*4 bytes) folds into
// the load's 24-bit immediate offset -> one address register per stream.
// Per K-block all B fragments load into distinct registers (one clause,
// progressive s_wait_loadcnt), then 2*TILES_N WMMAs issue against TILES_N
// independent accumulators (no back-to-back RAW on the XDL pipe).
// B1t/B2t are pre-transposed [NOUT, K].
// ---------------------------------------------------------------------------
template <bool HAS_A2, bool HAS_DEG, int TILES_N, int K>
__global__ void wmma_gemm_rowblock(const float* __restrict__ A1,
                                   const float* __restrict__ deg,
                                   const float* __restrict__ B1t,
                                   const float* __restrict__ A2,
                                   const float* __restrict__ B2t,
                                   const float* __restrict__ bias,
                                   float* __restrict__ out,
                                   int nrows) {
    constexpr int NOUT = TILES_N * 16;
    const int lane = threadIdx.x & 31;
    const int wave = threadIdx.x >> 5;
    const int rb = blockIdx.x * (blockDim.x >> 5) + wave;   // 16-row block
    if (rb * 16 >= nrows) return;                           // wave-uniform

    const int m  = lane & 15;    // row within block (A) / col within tile (B,C)
    const int hi = lane >> 4;    // K-pair select: lanes 16-31 hold K+2,K+3
    const long row = (long)rb * 16 + m;

    float scale = 1.0f;
    if (HAS_DEG) scale = 1.0f / fmaxf(deg[row], 1.0f);

    // single base pointer per stream; kb advances it, the column-tile index
    // is a compile-time immediate offset (tn * 16 * K floats)
    const float* ap1 = A1 + row * K + 2 * hi;
    const float* ap2 = HAS_A2 ? (A2 + row * K + 2 * hi) : nullptr;
    const float* bp1 = B1t + (long)m * K + 2 * hi;
    const float* bp2 = HAS_A2 ? (B2t + (long)m * K + 2 * hi) : nullptr;

    v8f acc[TILES_N];
#pragma unroll
    for (int tn = 0; tn < TILES_N; ++tn)
        acc[tn] = (v8f){0.f, 0.f, 0.f, 0.f, 0.f, 0.f, 0.f, 0.f};

#pragma unroll 1
    for (int kb = 0; kb < K; kb += 4) {
        v2f a1 = *(const v2f*)(ap1 + kb);
        if (HAS_DEG) a1 *= scale;                  // mean aggregation
        v2f a2;
        if (HAS_A2) a2 = *(const v2f*)(ap2 + kb);

        v2f b1[TILES_N];
        v2f b2[TILES_N];
#pragma unroll
        for (int tn = 0; tn < TILES_N; ++tn) {
            b1[tn] = *(const v2f*)(bp1 + kb + tn * 16 * K);
            if (HAS_A2) b2[tn] = *(const v2f*)(bp2 + kb + tn * 16 * K);
        }

#pragma unroll
        for (int tn = 0; tn < TILES_N; ++tn) {
            acc[tn] = __builtin_amdgcn_wmma_f32_16x16x4_f32(
                false, a1, false, b1[tn], (short)0, acc[tn], false, false);
            if (HAS_A2) {
                acc[tn] = __builtin_amdgcn_wmma_f32_16x16x4_f32(
                    false, a2, false, b2[tn], (short)0, acc[tn], false, false);
            }
        }
    }

#pragma unroll
    for (int tn = 0; tn < TILES_N; ++tn) {
        const int col = tn * 16 + m;
        const float bv = bias[col];
#pragma unroll
        for (int r = 0; r < 8; ++r) {
            float v = acc[tn][r] + bv;
            v = fmaxf(v, 0.0f);                    // every GEMM here is ReLU'd
            out[(long)(rb * 16 + r + 8 * hi) * NOUT + col] = v;
        }
    }
}

// ---------------------------------------------------------------------------
// In-degree count (float, matches jnp segment_sum of ones)
// ---------------------------------------------------------------------------
__global__ void degree_kernel(const int* __restrict__ dst, float* __restrict__ deg, int E_) {
    int i = blockIdx.x * blockDim.x + threadIdx.x;
    if (i < E_) atomicAdd(&deg[dst[i]], 1.0f);
}

// ---------------------------------------------------------------------------
// Edge gather + scatter-add (segment_sum of h[src] into agg[dst]).
// One wave32 per edge: lane l moves features [4l..4l+3] (512B coalesced row).
// ---------------------------------------------------------------------------
__global__ void scatter_kernel(const float* __restrict__ h,
                               const int* __restrict__ src,
                               const int* __restrict__ dst,
                               float* __restrict__ agg, int E_) {
    const int lane = threadIdx.x & 31;
    const int wave = threadIdx.x >> 5;
    const long e = (long)blockIdx.x * 8 + wave;
    if (e >= E_) return;
    const int s = src[e];
    const int d = dst[e];
    const float4 v = *(const float4*)&h[(long)s * 128 + lane * 4];
    float* base = &agg[(long)d * 128 + lane * 4];
    atomicAdd(base + 0, v.x);
    atomicAdd(base + 1, v.y);
    atomicAdd(base + 2, v.z);
    atomicAdd(base + 3, v.w);
}

// ---------------------------------------------------------------------------
// Per-node 2-way attention softmax + weighted sum. One wave32 per node.
// att_in = concat(out0,out1) [256]; w_att [256,2]; aggregated [N,128].
// ---------------------------------------------------------------------------
__global__ void att_aggregate_kernel(const float* __restrict__ out0,
                                     const float* __restrict__ out1,
                                     const float* __restrict__ w_att,
                                     const float* __restrict__ b_att,
                                     float* __restrict__ aggregated, int n) {
    const int lane = threadIdx.x & 31;
    const int wave = threadIdx.x >> 5;
    const int node = blockIdx.x * 8 + wave;
    if (node >= n) return;

    const float4 a = *(const float4*)&out0[(long)node * 128 + lane * 4];
    const float4 b = *(const float4*)&out1[(long)node * 128 + lane * 4];
    const float av[4] = {a.x, a.y, a.z, a.w};
    const float bv[4] = {b.x, b.y, b.z, b.w};

    float z0 = 0.f, z1 = 0.f;
#pragma unroll
    for (int j = 0; j < 4; ++j) {
        const int k = lane * 4 + j;
        z0 += av[j] * w_att[k * 2 + 0] + bv[j] * w_att[(k + 128) * 2 + 0];
        z1 += av[j] * w_att[k * 2 + 1] + bv[j] * w_att[(k + 128) * 2 + 1];
    }
#pragma unroll
    for (int off = 16; off > 0; off >>= 1) {
        z0 += __shfl_xor(z0, off, 32);
        z1 += __shfl_xor(z1, off, 32);
    }
    z0 += b_att[0];
    z1 += b_att[1];
    const float mx = fmaxf(z0, z1);
    const float e0 = expf(z0 - mx), e1 = expf(z1 - mx);
    const float inv = 1.0f / (e0 + e1);
    const float a0 = e0 * inv, a1 = e1 * inv;

    float4 r;
    r.x = a0 * a.x + a1 * b.x;
    r.y = a0 * a.y + a1 * b.y;
    r.z = a0 * a.z + a1 * b.z;
    r.w = a0 * a.w + a1 * b.w;
    *(float4*)&aggregated[(long)node * 128 + lane * 4] = r;
}

// ---------------------------------------------------------------------------
// Final tiny heads: fraud = hid_c@w_c2+b_c2 [N,2], pattern = hid_p@w_p2+b_p2 [N,5]
// One wave32 per node; lane l covers hidden indices l and l+32.
// ---------------------------------------------------------------------------
__global__ void heads_kernel(const float* __restrict__ hidc,
                             const float* __restrict__ hidp,
                             const float* __restrict__ w_c2, const float* __restrict__ b_c2,
                             const float* __restrict__ w_p2, const float* __restrict__ b_p2,
                             float* __restrict__ fraud, float* __restrict__ pat, int n) {
    const int lane = threadIdx.x & 31;
    const int wave = threadIdx.x >> 5;
    const int node = blockIdx.x * 8 + wave;
    if (node >= n) return;

    const float hc0 = hidc[(long)node * 64 + lane];
    const float hc1 = hidc[(long)node * 64 + 32 + lane];
    const float hp0 = hidp[(long)node * 64 + lane];
    const float hp1 = hidp[(long)node * 64 + 32 + lane];

    float acc[7];
#pragma unroll
    for (int cc = 0; cc < 2; ++cc)
        acc[cc] = hc0 * w_c2[lane * 2 + cc] + hc1 * w_c2[(lane + 32) * 2 + cc];
#pragma unroll
    for (int cc = 0; cc < 5; ++cc)
        acc[2 + cc] = hp0 * w_p2[lane * 5 + cc] + hp1 * w_p2[(lane + 32) * 5 + cc];

#pragma unroll
    for (int off = 16; off > 0; off >>= 1) {
#pragma unroll
        for (int i = 0; i < 7; ++i) acc[i] += __shfl_xor(acc[i], off, 32);
    }

    if (lane == 0) {
        fraud[(long)node * 2 + 0] = acc[0] + b_c2[0];
        fraud[(long)node * 2 + 1] = acc[1] + b_c2[1];
#pragma unroll
        for (int cc = 0; cc < 5; ++cc)
            pat[(long)node * 5 + cc] = acc[2 + cc] + b_p2[cc];
    }
}

// ---------------------------------------------------------------------------
extern "C" void kernel_launch(void* const* d_in, const int* in_sizes, int n_in,
                              void* d_out, int out_size, void* d_ws, size_t ws_size,
                              hipStream_t stream) {
    const float* x     = (const float*)d_in[0];
    const int*   ei0   = (const int*)d_in[1];
    const int*   ei1   = (const int*)d_in[2];
    const float* w_emb = (const float*)d_in[3];
    const float* b_emb = (const float*)d_in[4];
    const float* Wl    = (const float*)d_in[5];
    const float* bl    = (const float*)d_in[6];
    const float* Wr    = (const float*)d_in[7];
    const float* w_att = (const float*)d_in[8];
    const float* b_att = (const float*)d_in[9];
    const float* w_c1  = (const float*)d_in[10];
    const float* b_c1  = (const float*)d_in[11];
    const float* w_c2  = (const float*)d_in[12];
    const float* b_c2  = (const float*)d_in[13];
    const float* w_p1  = (const float*)d_in[14];
    const float* b_p1  = (const float*)d_in[15];
    const float* w_p2  = (const float*)d_in[16];
    const float* b_p2  = (const float*)d_in[17];

    const int H_ = 128;
    const int N_ = in_sizes[0] / H_;   // 100000 (multiple of 16)
    const int E_ = in_sizes[1] / 2;    // 1600000

    float* out        = (float*)d_out;
    float* fraud      = out;                       // [N,2]
    float* pat        = out + (size_t)N_ * 2;      // [N,5]
    float* aggregated = out + (size_t)N_ * 7;      // [N,128]

    float* w = (float*)d_ws;
    const size_t NH = (size_t)N_ * H_;
    float* h0   = w;                // [N,H]
    float* tmp1 = w + NH;           // ping
    float* tmp2 = w + 2 * NH;       // pong
    float* aggb = w + 3 * NH;       // scatter accumulator
    float* outs[2] = { w + 4 * NH, w + 5 * NH };
    float* degp[2] = { w + 6 * NH, w + 6 * NH + N_ };
    float* hidc = tmp1;             // reuse after conv phase [N,64]
    float* hidp = tmp2;             // reuse after conv phase [N,64]

    // transposed weights region (after deg arrays)
    float* wT = w + 6 * NH + 2 * (size_t)N_;
    const size_t HH = (size_t)H_ * H_;       // 16384
    float* w_embT = wT;                                  // [H,H]
    float* WlT    = w_embT + HH;                         // 6 x [H,H]
    float* WrT    = WlT + 6 * HH;                        // 6 x [H,H]
    float* w_c1T  = WrT + 6 * HH;                        // [64,H]
    float* w_p1T  = w_c1T + (size_t)64 * H_;             // [64,H]

    const int* eidx[2] = { ei0, ei1 };

    // --- one-time weight transposes (tiny) ---
    {
        const int TB = 256, NE = H_ * H_;
        transpose_kernel<<<(NE + TB - 1) / TB, TB, 0, stream>>>(w_emb, w_embT, H_, H_);
        for (int i = 0; i < 6; ++i) {
            transpose_kernel<<<(NE + TB - 1) / TB, TB, 0, stream>>>(
                Wl + (size_t)i * HH, WlT + (size_t)i * HH, H_, H_);
            transpose_kernel<<<(NE + TB - 1) / TB, TB, 0, stream>>>(
                Wr + (size_t)i * HH, WrT + (size_t)i * HH, H_, H_);
        }
        const int NE2 = H_ * 64;
        transpose_kernel<<<(NE2 + TB - 1) / TB, TB, 0, stream>>>(w_c1, w_c1T, H_, 64);
        transpose_kernel<<<(NE2 + TB - 1) / TB, TB, 0, stream>>>(w_p1, w_p1T, H_, 64);
    }

    // --- in-degrees (fixed per edge type across layers) ---
    hipMemsetAsync(degp[0], 0, 2 * (size_t)N_ * sizeof(float), stream);
    for (int et = 0; et < 2; ++et)
        degree_kernel<<<(E_ + 255) / 256, 256, 0, stream>>>(eidx[et] + E_, degp[et], E_);

    const int rowblocks = N_ / 16;                 // 6250
    const int gemm_grid = (rowblocks + 7) / 8;     // 8 waves / block

    // --- node embedding: h0 = relu(x @ w_emb + b_emb) ---
    wmma_gemm_rowblock<false, false, 8, 128><<<gemm_grid, 256, 0, stream>>>(
        x, nullptr, w_embT, nullptr, nullptr, b_emb, h0, N_);

    // --- 2 edge types x 3 SAGEConv layers ---
    for (int et = 0; et < 2; ++et) {
        const int* src = eidx[et];
        const int* dst = eidx[et] + E_;
        const float* hprev = h0;
        for (int l = 0; l < 3; ++l) {
            hipMemsetAsync(aggb, 0, NH * sizeof(float), stream);
            scatter_kernel<<<(E_ + 7) / 8, 256, 0, stream>>>(hprev, src, dst, aggb, E_);
            float* hout = (l == 2) ? outs[et] : ((l == 0) ? tmp1 : tmp2);
            const float* WlT_ = WlT + (size_t)(et * 3 + l) * HH;
            const float* WrT_ = WrT + (size_t)(et * 3 + l) * HH;
            const float* bl_  = bl + (size_t)(et * 3 + l) * H_;
            // relu( (agg/deg) @ Wl + bl + hprev @ Wr )
            wmma_gemm_rowblock<true, true, 8, 128><<<gemm_grid, 256, 0, stream>>>(
                aggb, degp[et], WlT_, hprev, WrT_, bl_, hout, N_);
            hprev = hout;
        }
    }

    // --- attention softmax + weighted aggregation -> d_out aggregated section ---
    att_aggregate_kernel<<<(N_ + 7) / 8, 256, 0, stream>>>(
        outs[0], outs[1], w_att, b_att, aggregated, N_);

    // --- head hidden layers (WMMA): relu(agg @ w_c1 + b_c1), relu(agg @ w_p1 + b_p1) ---
    wmma_gemm_rowblock<false, false, 4, 128><<<gemm_grid, 256, 0, stream>>>(
        aggregated, nullptr, w_c1T, nullptr, nullptr, b_c1, hidc, N_);
    wmma_gemm_rowblock<false, false, 4, 128><<<gemm_grid, 256, 0, stream>>>(
        aggregated, nullptr, w_p1T, nullptr, nullptr, b_p1, hidp, N_);

    // --- final tiny linears ---
    heads_kernel<<<(N_ + 7) / 8, 256, 0, stream>>>(
        hidc, hidp, w_c2, b_c2, w_p2, b_p2, fraud, pat, N_);
}